// RegularSparseLayer_51642686767265
// MI455X (gfx1250) — compile-verified
//
#include <hip/hip_runtime.h>

typedef __attribute__((ext_vector_type(2))) float v2f;
typedef __attribute__((ext_vector_type(8))) float v8f;

static constexpr int IN_F  = 8192;
static constexpr int OUT_F = 512;
static constexpr int SF    = 16;   // IN_F / OUT_F

// out[b, o] = sum_{k<16} x[b, 16*o + k] * w[o, 16*o + k] + bias[o]
//
// Wave tile: 16 batch rows (M) x 16 outputs (N), K = 256 columns of x
// consumed as 64 chained V_WMMA_F32_16X16X4_F32 steps. The banded weight
// contributes exactly one nonzero B-column per K-step (n = step/4), selected
// per-lane from a preloaded 8-float weight fragment.
__global__ __launch_bounds__(256, 2)
void banded_linear_wmma(const float* __restrict__ x,
                        const float* __restrict__ w,
                        const float* __restrict__ bias,
                        float* __restrict__ out)
{
    const int tid  = threadIdx.x;
    const int lane = tid & 31;
    const int wave = tid >> 5;
    const int ln   = lane & 15;      // N index (B/C/D) and M index (A)
    const int half = lane >> 4;      // 0: K={0,1} / M=r ; 1: K={2,3} / M=r+8
    const int kb   = half * 2;

    const int o0 = blockIdx.x * 16;                 // output tile base
    const int r0 = (blockIdx.y * 8 + wave) * 16;    // batch tile base
    const int c0 = o0 * SF;                         // x column base (16*o0)

    // Per-lane banded weight fragment: lane (half,ln) needs
    // w[o0+ln, c0 + 16*ln + 4q + kb + {0,1}] for q = 0..3  (8 floats).
    const float* wrow = w + (size_t)(o0 + ln) * IN_F + (c0 + 16 * ln + kb);
    v2f wf[4];
    #pragma unroll
    for (int q = 0; q < 4; ++q)
        wf[q] = *(const v2f*)(wrow + 4 * q);

    // A fragment source: lane (half,ln) reads x[r0+ln, c0 + 4s + kb + {0,1}]
    const float* xrow = x + (size_t)(r0 + ln) * IN_F + (c0 + kb);

    v8f acc = {};
    #pragma unroll
    for (int n = 0; n < 16; ++n) {          // active B column for this group
        const bool sel = (ln == n);
        #pragma unroll
        for (int q = 0; q < 4; ++q) {       // step s = 4n + q
            v2f a = *(const v2f*)(xrow + 16 * n + 4 * q);   // global_load_b64
            v2f zero = {0.0f, 0.0f};
            v2f b = sel ? wf[q] : zero;     // banded B: one nonzero column
            acc = __builtin_amdgcn_wmma_f32_16x16x4_f32(
                /*neg_a=*/false, a, /*neg_b=*/false, b,
                /*c_mod=*/(short)0, acc, /*reuse_a=*/false, /*reuse_b=*/false);
        }
    }

    // C/D layout: VGPR r -> row (r0 + r + 8*half), col (o0 + ln)
    const float bv = bias[o0 + ln];
    float* orow = out + (size_t)(r0 + half * 8) * OUT_F + (o0 + ln);
    #pragma unroll
    for (int r = 0; r < 8; ++r)
        orow[(size_t)r * OUT_F] = acc[r] + bv;
}

extern "C" void kernel_launch(void* const* d_in, const int* in_sizes, int n_in,
                              void* d_out, int out_size, void* d_ws, size_t ws_size,
                              hipStream_t stream) {
    const float* x    = (const float*)d_in[0];   // [4096, 8192] fp32
    const float* wgt  = (const float*)d_in[1];   // [512, 8192]  fp32
    const float* bias = (const float*)d_in[2];   // [512]        fp32
    float* out = (float*)d_out;                  // [4096, 512]  fp32

    const int batch = in_sizes[0] / IN_F;        // 4096
    dim3 grid(OUT_F / 16, batch / 128);          // 32 x 32 workgroups
    banded_linear_wmma<<<grid, dim3(256), 0, stream>>>(x, wgt, bias, out);
}